// Capsule_48842368090485
// MI455X (gfx1250) — compile-verified
//
#include <hip/hip_runtime.h>
#include <math.h>

// Problem constants (match reference)
#define BB       32      // batch
#define NCAP     64      // num_capsule
#define DCAP     32      // dim_capsule
#define INCAPS   2048    // in capsules
#define INDIM    16      // in dim (K)
#define INV_LOG2 1.4426950408889634f

typedef __attribute__((ext_vector_type(2))) float v2f;
typedef __attribute__((ext_vector_type(8))) float v8f;

// -----------------------------------------------------------------------------
// K1: u_hat[n,i,b,d] = sum_k W[n,i,d,k] * x[b,i,k]   (note layout: [N,In,B,D])
// One wave per (n,i): M=b (2 tiles of 16), N=d (2 tiles of 16),
// K=16 (4 steps of 4) via V_WMMA_F32_16X16X4_F32.
// Output tile for one (n,i) is a contiguous 4KB block -> stores use a single
// base address + immediate offsets (no per-store 64-bit address chains).
// -----------------------------------------------------------------------------
__global__ void uhat_gemm_kernel(const float* __restrict__ x,   // [B, In, 16]
                                 const float* __restrict__ W,   // [N, In, 32, 16]
                                 float* __restrict__ uhat)      // [N, In, B, 32]
{
    const int lane = threadIdx.x & 31;
    const int wave = threadIdx.x >> 5;            // 0..7
    const int i    = blockIdx.x * 8 + wave;       // in-capsule index
    const int n    = blockIdx.y;

    const int half = lane >> 4;                   // 0/1 -> k sub-pair
    const int l15  = lane & 15;

    // A fragments: X[b, i, k] (16x4 f32 layout)
    v2f afr[2][4];
#pragma unroll
    for (int tb = 0; tb < 2; ++tb) {
        const int b = tb * 16 + l15;
        const float* xp = x + ((size_t)b * INCAPS + i) * INDIM;
#pragma unroll
        for (int s = 0; s < 4; ++s) {
            const int k0 = 4 * s + 2 * half;      // even -> 8B aligned
            afr[tb][s] = *(const v2f*)(xp + k0);
        }
    }
    // B fragments: B[k, d] = W[n, i, d, k] (4x16 layout)
    v2f bfr[2][4];
#pragma unroll
    for (int td = 0; td < 2; ++td) {
        const int d = td * 16 + l15;
        const float* wp = W + (((size_t)n * INCAPS + i) * DCAP + d) * INDIM;
#pragma unroll
        for (int s = 0; s < 4; ++s) {
            const int k0 = 4 * s + 2 * half;
            bfr[td][s] = *(const v2f*)(wp + k0);
        }
    }

    v8f acc[2][2] = {};
#pragma unroll
    for (int tb = 0; tb < 2; ++tb)
#pragma unroll
        for (int td = 0; td < 2; ++td)
#pragma unroll
            for (int s = 0; s < 4; ++s)
                acc[tb][td] = __builtin_amdgcn_wmma_f32_16x16x4_f32(
                    /*neg_a=*/false, afr[tb][s],
                    /*neg_b=*/false, bfr[td][s],
                    /*c_mod=*/(short)0, acc[tb][td],
                    /*reuse_a=*/false, /*reuse_b=*/false);

    // Store: C tile (tb,td): VGPR v, lane -> b = tb*16 + v + 8*half, d = td*16+l15
    // One per-lane base pointer; every store offset is a compile-time constant.
    float* op = uhat + ((size_t)n * INCAPS + i) * (BB * DCAP)
                     + (8 * half) * DCAP + l15;
#pragma unroll
    for (int tb = 0; tb < 2; ++tb)
#pragma unroll
        for (int td = 0; td < 2; ++td)
#pragma unroll
            for (int v = 0; v < 8; ++v)
                op[(tb * 16 + v) * DCAP + td * 16] = acc[tb][td][v];
}

// -----------------------------------------------------------------------------
// K2 (fused): s[b,n,d] = sum_i c[b,n,i]*u_hat[n,i,b,d]; out = squash(s);
// then (mode>0) agreement b[b,n,i] (+)= sum_d u_hat[n,i,b,d]*out[b,n,d],
// re-reading the same 256KB slab while it is still hot in L2.
// mode: 0 = s+squash only, 1 = also b = agree, 2 = also b += agree.
// One block (256 threads = 8 waves) per (n,b).
// -----------------------------------------------------------------------------
__global__ void routing_fused_kernel(const float* __restrict__ uhat, // [N,In,B,32]
                                     const float* __restrict__ cij,  // [B,N,In] or null
                                     float* __restrict__ out,        // [B,N,32]
                                     float* __restrict__ bij,        // [B,N,In]
                                     int mode)
{
    const int n = blockIdx.x;
    const int b = blockIdx.y;
    const int d = threadIdx.x & 31;
    const int g = threadIdx.x >> 5;   // 8 waves

    const float* up = uhat + ((size_t)n * INCAPS * BB + b) * DCAP;
    const float* cp = cij ? (cij + ((size_t)b * NCAP + n) * INCAPS) : nullptr;

    float acc = 0.0f;
    for (int i = g; i < INCAPS; i += 8) {
        const float cv = cp ? cp[i] : (1.0f / 64.0f);
        acc += cv * up[(size_t)i * (BB * DCAP) + d];
    }

    __shared__ float red[8][32];
    __shared__ float o[DCAP];
    red[g][d] = acc;
    __syncthreads();

    if (g == 0) {
        float s = 0.0f;
#pragma unroll
        for (int q = 0; q < 8; ++q) s += red[q][d];
        // ||s||^2 across the 32 d-lanes (one wave32)
        float sq = s * s;
#pragma unroll
        for (int off = 16; off > 0; off >>= 1) sq += __shfl_xor(sq, off, 32);
        const float scale = sqrtf(sq) / (1.0f + sq);
        const float ov = scale * s;
        out[((size_t)b * NCAP + n) * DCAP + d] = ov;
        o[d] = ov;
    }
    if (mode == 0) return;          // uniform branch across the whole block
    __syncthreads();

    float* bp = bij + ((size_t)b * NCAP + n) * INCAPS;
    for (int i = threadIdx.x; i < INCAPS; i += 256) {
        const float4* u4 = (const float4*)(up + (size_t)i * (BB * DCAP));
        float a2 = 0.0f;
#pragma unroll
        for (int q = 0; q < 8; ++q) {
            const float4 u = u4[q];
            a2 += u.x * o[4 * q] + u.y * o[4 * q + 1] +
                  u.z * o[4 * q + 2] + u.w * o[4 * q + 3];
        }
        bp[i] = (mode == 2) ? (bp[i] + a2) : a2;
    }
}

// -----------------------------------------------------------------------------
// K3: c[b,:,i] = softmax_n(b[b,:,i] * INV_LOG2). One thread per (b,i);
// lanes map to consecutive i -> every n-row access is a coalesced 128B line.
// -----------------------------------------------------------------------------
__global__ void softmax_n_kernel(const float* __restrict__ bij,
                                 float* __restrict__ cij)
{
    const int b = blockIdx.y;
    const int i = blockIdx.x * blockDim.x + threadIdx.x;

    const float* bp = bij + (size_t)b * NCAP * INCAPS + i;
    float y[NCAP];
    float m = -INFINITY;
#pragma unroll
    for (int n = 0; n < NCAP; ++n) {
        y[n] = bp[(size_t)n * INCAPS] * INV_LOG2;
        m = fmaxf(m, y[n]);
    }
    float sum = 0.0f;
#pragma unroll
    for (int n = 0; n < NCAP; ++n) {
        y[n] = __expf(y[n] - m);
        sum += y[n];
    }
    const float inv = 1.0f / sum;
    float* cp = cij + (size_t)b * NCAP * INCAPS + i;
#pragma unroll
    for (int n = 0; n < NCAP; ++n) cp[(size_t)n * INCAPS] = y[n] * inv;
}

// -----------------------------------------------------------------------------
// Host side
// -----------------------------------------------------------------------------
extern "C" void kernel_launch(void* const* d_in, const int* in_sizes, int n_in,
                              void* d_out, int out_size, void* d_ws, size_t ws_size,
                              hipStream_t stream) {
    const float* x = (const float*)d_in[0];   // inputs [32,2048,16]
    const float* W = (const float*)d_in[1];   // W [64,2048,32,16]
    float* out = (float*)d_out;               // [32,64,32]

    // Workspace layout: u_hat (512 MB) | b (16 MB) | c (16 MB)  => needs 544 MB
    const size_t UHAT_ELEMS = (size_t)BB * NCAP * INCAPS * DCAP;
    const size_t BNI_ELEMS  = (size_t)BB * NCAP * INCAPS;
    char* ws = (char*)d_ws;
    float* uhat = (float*)ws;
    float* bij  = (float*)(ws + UHAT_ELEMS * sizeof(float));
    float* cij  = (float*)(ws + (UHAT_ELEMS + BNI_ELEMS) * sizeof(float));
    (void)ws_size; (void)in_sizes; (void)n_in; (void)out_size;

    // u_hat GEMM: one wave per (n,i); 8 waves per block
    uhat_gemm_kernel<<<dim3(INCAPS / 8, NCAP), 256, 0, stream>>>(x, W, uhat);

    const dim3 gBN(NCAP, BB);
    const dim3 gSM(INCAPS / 256, BB);

    // Routing iteration 0 (c uniform = 1/64), fused agreement (b = agree)
    routing_fused_kernel<<<gBN, 256, 0, stream>>>(uhat, nullptr, out, bij, 1);
    softmax_n_kernel<<<gSM, 256, 0, stream>>>(bij, cij);

    // Routing iteration 1, fused agreement (b += agree)
    routing_fused_kernel<<<gBN, 256, 0, stream>>>(uhat, cij, out, bij, 2);
    softmax_n_kernel<<<gSM, 256, 0, stream>>>(bij, cij);

    // Routing iteration 2 (final output only)
    routing_fused_kernel<<<gBN, 256, 0, stream>>>(uhat, cij, out, bij, 0);
}